// PolyModel_65841848648096
// MI455X (gfx1250) — compile-verified
//
#include <hip/hip_runtime.h>

// CDNA5 / gfx1250: wave32, WMMA f32 16x16x4 path.
typedef __attribute__((ext_vector_type(2))) float v2f;
typedef __attribute__((ext_vector_type(8))) float v8f;

#define SZ 128
#define LD 130   // padded LDS row stride (floats) to avoid 64-bank stride-128 conflicts
#define NSTEP 14 // X1..X14

__device__ __forceinline__ v8f wmma_f32_4(v2f a, v2f b, v8f c) {
  // 8 args: (neg_a, A, neg_b, B, c_mod, C, reuse_a, reuse_b)
  return __builtin_amdgcn_wmma_f32_16x16x4_f32(false, a, false, b, (short)0, c,
                                               false, false);
}

__global__ __launch_bounds__(256, 1)
void poly_chain_kernel(const float* __restrict__ X,
                       const float* __restrict__ a,
                       float* __restrict__ out) {
  __shared__ float buf0[SZ * LD];
  __shared__ float buf1[SZ * LD];

  const int tid  = threadIdx.x;
  const int wave = tid >> 5;      // 0..7 : tile-row owned by this wave
  const int lane = tid & 31;
  const int lr   = lane & 15;     // row (A) / col (B,C) within tile
  const int lh   = lane >> 4;     // half-select: K pair for A/B, M+8 for C/D

  __builtin_prefetch(X, 0, 0);

  // c = exp(0.5)*(2^(N/2)-1) - sum|a| ; N=14 -> 2^7-1 = 127
  float suma = 0.0f;
  for (int i = 0; i < NSTEP - 1; ++i) suma += fabsf(a[i]);
  const float cN = 1.64872127070012819416f * 127.0f - suma;

  const int tm = wave;            // this wave's M-tile index (0..7)
  v8f yacc[8];                    // Y accumulator: 8 tiles x 8 VGPRs = 64 VGPRs/lane

  // ---- Step 0: X1 = I - X * X^T  (A,B read straight from global/L2) ----
  {
    const float a0 = a[0];
    for (int tn = 0; tn < 8; ++tn) {
      v8f acc = {};
      for (int k0 = 0; k0 < SZ; k0 += 4) {
        const int kk = k0 + lh * 2;
        v2f av, bv;
        // A tile of X (row-major): lane -> row tm*16+lr, cols kk, kk+1
        av.x = X[(tm * 16 + lr) * SZ + kk];
        av.y = X[(tm * 16 + lr) * SZ + kk + 1];
        // B tile of X^T: B[k][n] = X[n][k] -> lane col n reads row tn*16+lr of X
        bv.x = X[(tn * 16 + lr) * SZ + kk];
        bv.y = X[(tn * 16 + lr) * SZ + kk + 1];
        acc = wmma_f32_4(av, bv, acc);
      }
      v8f yt;
#pragma unroll
      for (int r = 0; r < 8; ++r) {
        const int gm = tm * 16 + r + 8 * lh;
        const int gn = tn * 16 + lr;
        const float id = (gm == gn) ? 1.0f : 0.0f;
        const float x1 = id - acc[r];
        buf0[gm * LD + gn] = x1;      // X1 -> LDS ping buffer
        yt[r] = id + a0 * x1;         // Y = I + a[0]*X1 + ...
      }
      yacc[tn] = yt;
    }
  }
  __syncthreads();

  // ---- Steps 1..13: repeated squaring in LDS, fold into Y on the fly ----
  for (int s = 1; s <= NSTEP - 1; ++s) {
    const float* src = (s & 1) ? buf0 : buf1;
    float*       dst = (s & 1) ? buf1 : buf0;
    const float  cf  = (s == NSTEP - 1) ? cN : a[s];
    for (int tn = 0; tn < 8; ++tn) {
      v8f acc = {};
      for (int k0 = 0; k0 < SZ; k0 += 4) {
        const int kk = k0 + lh * 2;
        v2f av, bv;
        av.x = src[(tm * 16 + lr) * LD + kk];
        av.y = src[(tm * 16 + lr) * LD + kk + 1];
        bv.x = src[kk * LD + tn * 16 + lr];
        bv.y = src[(kk + 1) * LD + tn * 16 + lr];
        acc = wmma_f32_4(av, bv, acc);
      }
#pragma unroll
      for (int r = 0; r < 8; ++r) {
        const int gm = tm * 16 + r + 8 * lh;
        const int gn = tn * 16 + lr;
        dst[gm * LD + gn] = acc[r];
        yacc[tn][r] += cf * acc[r];
      }
    }
    __syncthreads();
  }

  // ---- Spill Y (registers) to LDS for the final matmul ----
  for (int tn = 0; tn < 8; ++tn) {
#pragma unroll
    for (int r = 0; r < 8; ++r) {
      const int gm = tm * 16 + r + 8 * lh;
      const int gn = tn * 16 + lr;
      buf0[gm * LD + gn] = yacc[tn][r];
    }
  }
  __syncthreads();

  // ---- out = Y @ X  (A from LDS, B columns of X from global/L2) ----
  for (int tn = 0; tn < 8; ++tn) {
    v8f acc = {};
    for (int k0 = 0; k0 < SZ; k0 += 4) {
      const int kk = k0 + lh * 2;
      v2f av, bv;
      av.x = buf0[(tm * 16 + lr) * LD + kk];
      av.y = buf0[(tm * 16 + lr) * LD + kk + 1];
      bv.x = X[kk * SZ + tn * 16 + lr];
      bv.y = X[(kk + 1) * SZ + tn * 16 + lr];
      acc = wmma_f32_4(av, bv, acc);
    }
#pragma unroll
    for (int r = 0; r < 8; ++r) {
      const int gm = tm * 16 + r + 8 * lh;
      const int gn = tn * 16 + lr;
      out[gm * SZ + gn] = acc[r];
    }
  }
}

extern "C" void kernel_launch(void* const* d_in, const int* in_sizes, int n_in,
                              void* d_out, int out_size, void* d_ws, size_t ws_size,
                              hipStream_t stream) {
  const float* X = (const float*)d_in[0];   // [128*128] fp32
  const float* a = (const float*)d_in[1];   // [13] fp32
  float* out = (float*)d_out;               // [128*128] fp32
  poly_chain_kernel<<<dim3(1), dim3(256), 0, stream>>>(X, a, out);
}